// NonLinear_58557584114181
// MI455X (gfx1250) — compile-verified
//
#include <hip/hip_runtime.h>
#include <cstdint>

typedef __attribute__((ext_vector_type(4))) float        f32x4;
typedef __attribute__((ext_vector_type(2))) float        f32x2;
typedef __attribute__((ext_vector_type(4))) unsigned int u32x4;
typedef __attribute__((ext_vector_type(8))) unsigned int u32x8;

#define HW_ELEMS   (512 * 512)   // 2^18 elements per (n,c) plane
#define CCH        32
#define KNOTS      33
#define BLOCK      256
#define F4_PER_THR 8
#define CHUNK_F4   (BLOCK * F4_PER_THR)  // 2048 float4 per block
#define CHUNK_E    (CHUNK_F4 * 4)        // 8192 elements per block

__global__ __launch_bounds__(BLOCK)
void pls_shrink_kernel(const float* __restrict__ r,
                       const float* __restrict__ ps,
                       const float* __restrict__ qs,
                       float* __restrict__ out) {
    __shared__ float qraw[KNOTS * CCH];   // raw qs[K][C], 4224 B, filled by TDM
    __shared__ f32x2 tab[KNOTS - 1];      // per-channel {q_k, slope_k}, 256 B

    const int       tid   = threadIdx.x;
    const long long baseE = (long long)blockIdx.x * CHUNK_E;
    const int       c     = (int)((baseE >> 18) & (CCH - 1)); // whole block in one channel

    // ps is uniform linspace(-1,1,33): dp = 0.0625 exactly, so 1/dp and
    // p_k = fma(k, dp, p0) are bit-exact.
    const float p0     = ps[0];
    const float dp     = ps[1] - p0;
    const float pN     = ps[KNOTS - 1];
    const float inv_dp = 1.0f / dp;

    // --- Stage qs[K][C] into LDS via the Tensor Data Mover (wave 0 only) ---
    if (tid < warpSize) {
        const uint64_t ga      = (uint64_t)(uintptr_t)qs;
        const unsigned lds_off = (unsigned)(uintptr_t)&qraw[0]; // low 32 bits = LDS offset
        const unsigned NE      = KNOTS * CCH;                   // 1056 fp32 elements

        u32x4 g0;
        g0[0] = 0x1u;                                   // count=1 (valid user descriptor)
        g0[1] = lds_off;                                // lds_addr (bits 63:32)
        g0[2] = (unsigned)(ga & 0xFFFFFFFFull);         // global_addr[31:0]
        g0[3] = (unsigned)((ga >> 32) & 0x01FFFFFFull)  // global_addr[56:32]
              | 0x80000000u;                            // type=2 ("image")

        u32x8 g1;
        g1[0] = 0x2u << 16;               // workgroup_mask=0, data_size=2 (4 bytes)
        g1[1] = (NE & 0xFFFFu) << 16;     // tensor_dim0[15:0]  (bits 79:64... -> 63:48 of g1)
        g1[2] = (NE >> 16) | (1u << 16);  // tensor_dim0[31:16] | tensor_dim1 = 1
        g1[3] = (NE & 0xFFFFu) << 16;     // tile_dim0 = 1056 (bits 127:112)
        g1[4] = 0u;                       // tile_dim1 = tile_dim2 = 0 (1-D tile)
        g1[5] = NE;                       // tensor_dim0_stride[31:0]
        g1[6] = 0u;                       // stride hi / tensor_dim1_stride
        g1[7] = 0u;

        asm volatile("tensor_load_to_lds %0, %1" :: "s"(g0), "s"(g1) : "memory");
        __builtin_amdgcn_s_wait_tensorcnt(0);
    }
    __syncthreads();

    // --- Build per-channel {q_k, slope_k} pairs (one ds_load_b64 gather later) ---
    if (tid < KNOTS - 1) {
        const float qk  = qraw[tid * CCH + c];
        const float qk1 = qraw[(tid + 1) * CCH + c];
        f32x2 e;
        e.x = qk;
        e.y = (qk1 - qk) * inv_dp;
        tab[tid] = e;
    }
    __syncthreads();

    const float q0     = qraw[0 * CCH + c];           // LDS broadcast reads
    const float qN     = qraw[(KNOTS - 1) * CCH + c];
    const float lo_off = q0 - p0;                     // out_min = x + (q0 - p0)
    const float hi_off = qN - pN;                     // out_max = x + (qN - pN)

    const f32x4* rp = (const f32x4*)r   + (baseE >> 2) + tid;
    f32x4*       op = (f32x4*)out       + (baseE >> 2) + tid;

#pragma unroll
    for (int it = 0; it < F4_PER_THR; ++it) {
        f32x4 v = __builtin_nontemporal_load(rp + it * BLOCK); // global_load_b128 NT
        f32x4 o;
#pragma unroll
        for (int j = 0; j < 4; ++j) {
            const float x  = v[j];
            const float kf = floorf((x - p0) * inv_dp);
            const int   k  = (int)fminf(fmaxf(kf, 0.0f), (float)(KNOTS - 2));
            const f32x2 e  = tab[k];                       // ds_load_b64 gather
            const float pk  = fmaf((float)k, dp, p0);      // exact: dp = 2^-4
            const float mid = fmaf(e.y, x - pk, e.x);
            o[j] = (kf < 0.0f)                 ? (x + lo_off)
                 : (kf >= (float)(KNOTS - 1))  ? (x + hi_off)
                 : mid;
        }
        __builtin_nontemporal_store(o, op + it * BLOCK);   // global_store_b128 NT
    }
}

extern "C" void kernel_launch(void* const* d_in, const int* in_sizes, int n_in,
                              void* d_out, int out_size, void* d_ws, size_t ws_size,
                              hipStream_t stream) {
    const float* r  = (const float*)d_in[0];
    const float* ps = (const float*)d_in[1];
    const float* qs = (const float*)d_in[2];
    float*       o  = (float*)d_out;

    const int total = in_sizes[0];           // 67,108,864 elements
    const int grid  = total / CHUNK_E;       // 8192 blocks, each within one channel

    hipLaunchKernelGGL(pls_shrink_kernel, dim3(grid), dim3(BLOCK), 0, stream,
                       r, ps, qs, o);
}